// GraphEncoder_65773129171088
// MI455X (gfx1250) — compile-verified
//
#include <hip/hip_runtime.h>
#include <hip/hip_bf16.h>

typedef __attribute__((ext_vector_type(2))) float v2f;
typedef __attribute__((ext_vector_type(8))) float v8f;

// ---------------------------------------------------------------------------
// Normalization: deg = 1 (self loop) + sum over edges into dst; dinv = rsqrt.
// ---------------------------------------------------------------------------
__global__ void init_deg_kernel(float* deg, int N) {
    int i = blockIdx.x * blockDim.x + threadIdx.x;
    if (i < N) deg[i] = 1.0f;   // self-loop contribution
}

__global__ void degree_kernel(const int* __restrict__ dst, float* deg, int E) {
    int e = blockIdx.x * blockDim.x + threadIdx.x;
    if (e < E) atomicAdd(&deg[dst[e]], 1.0f);
}

__global__ void dinv_kernel(float* deg, int N) {
    int i = blockIdx.x * blockDim.x + threadIdx.x;
    if (i < N) deg[i] = rsqrtf(fmaxf(deg[i], 1.0f));   // in place -> dinv
}

__global__ void coef_kernel(const int* __restrict__ src, const int* __restrict__ dst,
                            const float* __restrict__ dinv, float* __restrict__ coef, int E) {
    int e = blockIdx.x * blockDim.x + threadIdx.x;
    if (e < E) coef[e] = dinv[src[e]] * dinv[dst[e]];
}

// ---------------------------------------------------------------------------
// GEMM  Out[N,64] = A[N,64] @ W[64,64]   via V_WMMA_F32_16X16X4_F32.
// One wave (32 lanes) per 16-row tile; 4 column tiles of 16; K loop 64/4.
// Fragment layouts per CDNA5 ISA 7.12.2:
//   A 16x4 : lanes 0-15 -> M=lane, K={k0,k0+1}; lanes 16-31 -> M=lane-16, K={k0+2,k0+3}
//   B 4x16 : lanes 0-15 -> rows K={k0,k0+1} at N=lane; lanes 16-31 -> rows {k0+2,k0+3}
//   C/D    : VGPR r, lanes 0-15 -> M=r, N=lane; lanes 16-31 -> M=r+8, N=lane-16
// ---------------------------------------------------------------------------
__global__ void gemm_wmma_f32(const float* __restrict__ A, const float* __restrict__ W,
                              float* __restrict__ Out) {
    const int tile = blockIdx.x;          // 16 output rows
    const int lane = threadIdx.x;         // 0..31, EXEC all ones
    const int m     = lane & 15;
    const int khalf = lane >> 4;          // 0: K pair {0,1}; 1: K pair {2,3}

    v8f acc0 = {}, acc1 = {}, acc2 = {}, acc3 = {};
    const float* __restrict__ Arow = A + (size_t)(tile * 16 + m) * 64;

    #pragma unroll
    for (int k0 = 0; k0 < 64; k0 += 4) {
        const int ka = k0 + khalf * 2;
        v2f a;
        a.x = Arow[ka];
        a.y = Arow[ka + 1];

        const float* __restrict__ Wr0 = W + (size_t)ka * 64;
        const float* __restrict__ Wr1 = Wr0 + 64;
        v2f b0, b1, b2, b3;
        b0.x = Wr0[m];      b0.y = Wr1[m];
        b1.x = Wr0[16 + m]; b1.y = Wr1[16 + m];
        b2.x = Wr0[32 + m]; b2.y = Wr1[32 + m];
        b3.x = Wr0[48 + m]; b3.y = Wr1[48 + m];

        acc0 = __builtin_amdgcn_wmma_f32_16x16x4_f32(false, a, false, b0, (short)0, acc0, false, false);
        acc1 = __builtin_amdgcn_wmma_f32_16x16x4_f32(false, a, false, b1, (short)0, acc1, false, false);
        acc2 = __builtin_amdgcn_wmma_f32_16x16x4_f32(false, a, false, b2, (short)0, acc2, false, false);
        acc3 = __builtin_amdgcn_wmma_f32_16x16x4_f32(false, a, false, b3, (short)0, acc3, false, false);
    }

    #pragma unroll
    for (int r = 0; r < 8; ++r) {
        const int mo = r + khalf * 8;
        float* __restrict__ orow = Out + (size_t)(tile * 16 + mo) * 64 + m;
        orow[0]  = acc0[r];
        orow[16] = acc1[r];
        orow[32] = acc2[r];
        orow[48] = acc3[r];
    }
}

// Scalar tail for N not multiple of 16 (not launched when N % 16 == 0).
__global__ void gemm_tail(const float* __restrict__ A, const float* __restrict__ W,
                          float* __restrict__ Out, int row0, int N) {
    int idx = blockIdx.x * blockDim.x + threadIdx.x;
    int row = row0 + (idx >> 6);
    int col = idx & 63;
    if (row >= N) return;
    float s = 0.0f;
    #pragma unroll
    for (int k = 0; k < 64; ++k) s += A[(size_t)row * 64 + k] * W[(size_t)k * 64 + col];
    Out[(size_t)row * 64 + col] = s;
}

// ---------------------------------------------------------------------------
// Aggregation: out[i] starts at self-loop term htmp[i]*dinv^2, then edges
// atomically add htmp[src]*coef into out[dst]; finally bias + ReLU.
// ---------------------------------------------------------------------------
__global__ void self_loop_init(const float* __restrict__ htmp, const float* __restrict__ dinv,
                               float* __restrict__ out, int N) {
    int i = blockIdx.x * blockDim.x + threadIdx.x;
    if (i >= N * 64) return;
    const float di = dinv[i >> 6];
    out[i] = htmp[i] * di * di;
}

__global__ void scatter_edges(const int* __restrict__ src, const int* __restrict__ dst,
                              const float* __restrict__ coef, const float* __restrict__ h,
                              float* __restrict__ out, int E) {
    int idx = blockIdx.x * blockDim.x + threadIdx.x;
    if (idx >= E * 16) return;
    const int e  = idx >> 4;
    const int c4 = (idx & 15) << 2;             // 4-float column chunk
    const int s  = src[e];
    const int d  = dst[e];
    const float cf = coef[e];
    const float4 v = *reinterpret_cast<const float4*>(h + (size_t)s * 64 + c4);
    float* op = out + (size_t)d * 64 + c4;
    atomicAdd(op + 0, v.x * cf);
    atomicAdd(op + 1, v.y * cf);
    atomicAdd(op + 2, v.z * cf);
    atomicAdd(op + 3, v.w * cf);
}

__global__ void bias_relu(float* __restrict__ h, const float* __restrict__ b, int N) {
    int i = blockIdx.x * blockDim.x + threadIdx.x;
    if (i >= N * 64) return;
    h[i] = fmaxf(h[i] + b[i & 63], 0.0f);
}

// ---------------------------------------------------------------------------
// Mean pooling over batch segments.
// ---------------------------------------------------------------------------
__global__ void zero_kernel(float* p, int n) {
    int i = blockIdx.x * blockDim.x + threadIdx.x;
    if (i < n) p[i] = 0.0f;
}

__global__ void pool_accum(const float* __restrict__ h, const int* __restrict__ batch,
                           float* __restrict__ pool, float* __restrict__ cnt, int N) {
    int i = blockIdx.x * blockDim.x + threadIdx.x;
    if (i >= N * 64) return;
    const int node = i >> 6;
    const int c    = i & 63;
    const int g    = batch[node];
    atomicAdd(&pool[(size_t)g * 64 + c], h[i]);
    if (c == 0) atomicAdd(&cnt[g], 1.0f);
}

__global__ void pool_div(const float* __restrict__ pool, const float* __restrict__ cnt,
                         float* __restrict__ out, int total) {
    int i = blockIdx.x * blockDim.x + threadIdx.x;
    if (i >= total) return;
    out[i] = pool[i] / fmaxf(cnt[i >> 6], 1.0f);
}

// ---------------------------------------------------------------------------
static inline size_t align16f(size_t nfloats) { return (nfloats + 3) & ~(size_t)3; }

extern "C" void kernel_launch(void* const* d_in, const int* in_sizes, int n_in,
                              void* d_out, int out_size, void* d_ws, size_t ws_size,
                              hipStream_t stream) {
    const float* x     = (const float*)d_in[0];
    const int*   eidx  = (const int*)d_in[1];      // [2, E] row-major: src then dst
    const int*   batch = (const int*)d_in[2];
    const float* W0    = (const float*)d_in[3];
    const float* b0    = (const float*)d_in[4];
    const float* W1    = (const float*)d_in[5];
    const float* b1    = (const float*)d_in[6];
    const float* W2    = (const float*)d_in[7];
    const float* b2    = (const float*)d_in[8];

    const int N = in_sizes[0] / 64;
    const int E = in_sizes[1] / 2;
    const int G = out_size / 64;

    const int* esrc = eidx;
    const int* edst = eidx + E;

    // Workspace layout (floats, 16B-aligned sections)
    float* wsf  = (float*)d_ws;
    size_t off  = 0;
    float* dinv = wsf + off; off += align16f((size_t)N);
    float* coef = wsf + off; off += align16f((size_t)E);
    float* htmp = wsf + off; off += align16f((size_t)N * 64);
    float* bufA = wsf + off; off += align16f((size_t)N * 64);
    float* bufB = wsf + off; off += align16f((size_t)N * 64);
    float* pool = wsf + off; off += align16f((size_t)G * 64);
    float* cnt  = wsf + off; off += align16f((size_t)G);
    (void)ws_size;

    const int T = 256;
    const int NE_ELEM = N * 64;

    // --- normalization ---
    init_deg_kernel<<<(N + T - 1) / T, T, 0, stream>>>(dinv, N);
    degree_kernel<<<(E + T - 1) / T, T, 0, stream>>>(edst, dinv, E);
    dinv_kernel<<<(N + T - 1) / T, T, 0, stream>>>(dinv, N);
    coef_kernel<<<(E + T - 1) / T, T, 0, stream>>>(esrc, edst, dinv, coef, E);

    const int tiles = N / 16;
    const int rem   = N - tiles * 16;

    const float* layerW[3] = {W0, W1, W2};
    const float* layerB[3] = {b0, b1, b2};
    const float* hin  = x;
    float*       houts[3] = {bufA, bufB, bufA};

    for (int l = 0; l < 3; ++l) {
        float* hout = houts[l];
        // h @ W  (tensor-core path)
        if (tiles > 0)
            gemm_wmma_f32<<<tiles, 32, 0, stream>>>(hin, layerW[l], htmp);
        if (rem > 0)
            gemm_tail<<<(rem * 64 + T - 1) / T, T, 0, stream>>>(hin, layerW[l], htmp, tiles * 16, N);
        // self-loop init (plain stores, no atomics needed before scatter)
        self_loop_init<<<(NE_ELEM + T - 1) / T, T, 0, stream>>>(htmp, dinv, hout, N);
        // edge scatter-add (L2-resident atomics)
        scatter_edges<<<(E * 16 + T - 1) / T, T, 0, stream>>>(esrc, edst, coef, htmp, hout, E);
        // bias + ReLU
        bias_relu<<<(NE_ELEM + T - 1) / T, T, 0, stream>>>(hout, layerB[l], N);
        hin = hout;
    }

    // --- global mean pool ---
    zero_kernel<<<(G * 64 + G + T - 1) / T, T, 0, stream>>>(pool, G * 64 + G);  // pool and cnt are contiguous
    pool_accum<<<(NE_ELEM + T - 1) / T, T, 0, stream>>>(hin, batch, pool, cnt, N);
    pool_div<<<(out_size + T - 1) / T, T, 0, stream>>>(pool, cnt, (float*)d_out, out_size);
}